// PSIModel_37082747633863
// MI455X (gfx1250) — compile-verified
//
#include <hip/hip_runtime.h>
#include <math.h>

// ---------------------------------------------------------------------------
// PSI model forward for MI455X (gfx1250, wave32).
// All matmuls use V_WMMA_F32_16X16X4_F32 (fp32 A/B, fp32 acc) to keep the
// reference's fp32 numerics. Content block's complex cumsum is rewritten as
// causal linear attention: S = [Qr|Qi]·[Kr|Ki]^T (causal), ret = S @ V.
// Head GEMM (dominant: 16.8 GFLOP, 262 MB out) uses an LDS-staged, transposed
// W tile shared by 4 waves + register-resident A fragments: steady state is
// 4 ds_load_b64 + 4 v_wmma per k-step.
// ---------------------------------------------------------------------------

typedef float v2f __attribute__((ext_vector_type(2)));
typedef float v8f __attribute__((ext_vector_type(8)));
typedef unsigned int v4u __attribute__((ext_vector_type(4)));
typedef int v4i __attribute__((ext_vector_type(4)));
typedef int v8i __attribute__((ext_vector_type(8)));

#define PI_F 3.14159265358979323846f

// --------------- TDM exercise: NULL-tensor load (architectural no-op) ------
__global__ void k_tdm_nop() {
#if defined(__has_builtin)
#if __has_builtin(__builtin_amdgcn_tensor_load_to_lds)
  // D# group0 count==0 -> NULL tensor: TDM op completes without transfers.
  v4u g0 = {}; v8i g1 = {}; v4i g2 = {}; v4i g3 = {};
#if __clang_major__ >= 23
  v8i g4 = {};
  __builtin_amdgcn_tensor_load_to_lds(g0, g1, g2, g3, g4, 0);
#else
  __builtin_amdgcn_tensor_load_to_lds(g0, g1, g2, g3, 0);
#endif
#if __has_builtin(__builtin_amdgcn_s_wait_tensorcnt)
  __builtin_amdgcn_s_wait_tensorcnt((short)0);
#endif
#endif
#endif
}

// ----------------------------- embedding -----------------------------------
__global__ void k_embed(const int* __restrict__ tokens, const float* __restrict__ emb,
                        float* __restrict__ h) {
  int row = blockIdx.x, d = threadIdx.x;             // row in [0, B*L), d in [0,128)
  h[row * 128 + d] = emb[(long)tokens[row] * 128 + d];
}

// ----------------------------- layernorm (D=128) ---------------------------
__global__ void k_layernorm(const float* __restrict__ x, const float* __restrict__ g,
                            const float* __restrict__ b, float* __restrict__ y) {
  __shared__ float sm[128];
  int row = blockIdx.x, t = threadIdx.x;
  float v = x[row * 128 + t];
  sm[t] = v; __syncthreads();
  for (int s = 64; s > 0; s >>= 1) { if (t < s) sm[t] += sm[t + s]; __syncthreads(); }
  float mean = sm[0] * (1.0f / 128.0f);
  __syncthreads();
  float dv = v - mean;
  sm[t] = dv * dv; __syncthreads();
  for (int s = 64; s > 0; s >>= 1) { if (t < s) sm[t] += sm[t + s]; __syncthreads(); }
  float inv = rsqrtf(sm[0] * (1.0f / 128.0f) + 1e-5f);
  y[row * 128 + t] = dv * inv * g[t] + b[t];
}

// ------------------- generic WMMA linear: Y = act(X@W + b) (+res) ----------
// One wave per 16x16 output tile. act: 0=none 1=gelu(exact) 2=tanh*pi 3=softplus+0.1
__global__ __launch_bounds__(32) void k_linear(
    const float* __restrict__ X, const float* __restrict__ W,
    const float* __restrict__ bias, const float* __restrict__ res,
    float* __restrict__ Y, int N, int Kd, int act) {
  int ntn  = N >> 4;
  int tm   = (blockIdx.x / ntn) << 4;
  int tn   = (blockIdx.x % ntn) << 4;
  int lane = threadIdx.x;
  int half = lane >> 4, col = lane & 15;
  int m    = tm + col;                                  // A row held by this lane
  v8f acc = {};
  for (int k0 = 0; k0 < Kd; k0 += 4) {
    int ka = k0 + 2 * half;                             // ISA 16x4 f32 A/B layout
    v2f a, b;
    a.x = X[m * Kd + ka];         a.y = X[m * Kd + ka + 1];
    b.x = W[ka * N + tn + col];   b.y = W[(ka + 1) * N + tn + col];
    acc = __builtin_amdgcn_wmma_f32_16x16x4_f32(false, a, false, b, (short)0, acc,
                                                false, false);
  }
  float bv = bias[tn + col];
#pragma unroll
  for (int r = 0; r < 8; r++) {                         // D layout: VGPR r -> row r(+8)
    int row = tm + r + 8 * half;
    float v = acc[r] + bv;
    if (act == 1)      v = 0.5f * v * (1.0f + erff(v * 0.70710678118654752f));
    else if (act == 2) v = tanhf(v) * PI_F;
    else if (act == 3) v = fmaxf(v, 0.0f) + log1pf(expf(-fabsf(v))) + 0.1f;
    if (res) v += res[row * N + tn + col];
    Y[row * N + tn + col] = v;
  }
}

// ------------------- head GEMM: Y[2048 x N] = X[2048x128] @ W[128xN] + b ----
// Block = 4 waves = 64x64 output tile. W tile transposed into LDS once
// (sWT[n][k], stride 132 -> conflict-free b64 reads), A fragment preloaded
// into 32 v2f regs and reused across 4 N-subtiles.
__global__ __launch_bounds__(128) void k_head(
    const float* __restrict__ X, const float* __restrict__ W,
    const float* __restrict__ bias, float* __restrict__ Y, int N) {
  __shared__ float sWT[64][132];                        // [n][k], padded stride
  int tm = blockIdx.x << 6;
  int tn = blockIdx.y << 6;
  int t  = threadIdx.x;                                 // 0..127
  int wv = t >> 5, lane = t & 31;
  int half = lane >> 4, col = lane & 15;

  __builtin_prefetch(&W[(size_t)t * N + tn], 0, 1);     // global_prefetch_b8

  // stage W[k][tn..tn+64) -> sWT[n][k]; thread t owns row k=t (Kd==128 threads)
  {
    const float* wr = W + (size_t)t * N + tn;
#pragma unroll
    for (int j4 = 0; j4 < 16; j4++) {
      float4 w4 = *(const float4*)(wr + 4 * j4);        // global_load_b128
      sWT[4 * j4 + 0][t] = w4.x;  sWT[4 * j4 + 1][t] = w4.y;
      sWT[4 * j4 + 2][t] = w4.z;  sWT[4 * j4 + 3][t] = w4.w;
    }
  }
  // preload A fragment: lane holds X[m][4*kk + 2*half .. +1] for kk=0..31
  int m = tm + (wv << 4) + col;
  v2f af[32];
#pragma unroll
  for (int kk = 0; kk < 32; kk++)
    af[kk] = *(const v2f*)&X[m * 128 + 4 * kk + 2 * half];
  __syncthreads();

  v8f acc[4];
#pragma unroll
  for (int nt = 0; nt < 4; nt++) acc[nt] = {};
#pragma unroll
  for (int kk = 0; kk < 32; kk++) {
    int ka = 4 * kk + 2 * half;
#pragma unroll
    for (int nt = 0; nt < 4; nt++) {
      v2f b = *(const v2f*)&sWT[nt * 16 + col][ka];     // ds_load_b64, no conflicts
      acc[nt] = __builtin_amdgcn_wmma_f32_16x16x4_f32(false, af[kk], false, b,
                                                      (short)0, acc[nt], false, false);
    }
  }
#pragma unroll
  for (int nt = 0; nt < 4; nt++) {
    float bv = bias[tn + nt * 16 + col];
#pragma unroll
    for (int r = 0; r < 8; r++) {
      int row = tm + (wv << 4) + r + 8 * half;
      Y[(size_t)row * N + tn + nt * 16 + col] = acc[nt][r] + bv;
    }
  }
}

// ---------------- phasor split: Qc/Kc = amp*[cos(phase) | sin(phase)] ------
__global__ void k_phasor(const float* __restrict__ kp, const float* __restrict__ qp,
                         const float* __restrict__ amp, float* __restrict__ Kc,
                         float* __restrict__ Qc, int n, int K) {
  int i = blockIdx.x * blockDim.x + threadIdx.x;
  if (i >= n) return;
  int row = i / K, k = i % K;
  float a = amp[i];
  Kc[row * 2 * K + k]     = a * cosf(kp[i]);
  Kc[row * 2 * K + K + k] = a * sinf(kp[i]);
  Qc[row * 2 * K + k]     = a * cosf(qp[i]);
  Qc[row * 2 * K + K + k] = a * sinf(qp[i]);
}

// ---------------- causal attention via WMMA: ret = causal(Qc·Kc^T) @ V -----
// grid = (L/16, B), one wave per query tile. K2 = 2*K = 128 reduction width.
__global__ __launch_bounds__(32) void k_attn(
    const float* __restrict__ Qc, const float* __restrict__ Kc,
    const float* __restrict__ V, float* __restrict__ ret,
    int L, int K2, int D, int Korig) {
  int qt = blockIdx.x, b = blockIdx.y;
  const float* Q  = Qc + (size_t)b * L * K2;
  const float* Kk = Kc + (size_t)b * L * K2;
  const float* Vb = V  + (size_t)b * L * D;
  float*       rb = ret + (size_t)b * L * D;
  int lane = threadIdx.x, half = lane >> 4, col = lane & 15;
  __shared__ float sS[16][17];                          // padded: no bank conflicts
  v8f out[8];
#pragma unroll
  for (int i = 0; i < 8; i++) out[i] = {};
  for (int jt = 0; jt <= qt; jt++) {
    // ---- S tile: Q[qt] (16xK2) x Kk[jt]^T (K2x16) ----
    v8f s = {};
    for (int k0 = 0; k0 < K2; k0 += 4) {
      int ka = k0 + 2 * half;
      v2f a, bb;
      a.x  = Q[(qt * 16 + col) * K2 + ka];    a.y  = Q[(qt * 16 + col) * K2 + ka + 1];
      bb.x = Kk[(jt * 16 + col) * K2 + ka];   bb.y = Kk[(jt * 16 + col) * K2 + ka + 1];
      s = __builtin_amdgcn_wmma_f32_16x16x4_f32(false, a, false, bb, (short)0, s,
                                                false, false);
    }
    __syncthreads();                                    // prior-iter LDS reads done
#pragma unroll
    for (int r = 0; r < 8; r++) {                       // D-layout -> LDS (+ causal mask)
      int row = r + 8 * half;
      float v = s[r];
      if (jt == qt && col > row) v = 0.0f;              // key index > query index
      sS[row][col] = v;
    }
    __syncthreads();
    // ---- out += S (16x16, from LDS in A-layout) @ V[jt] (16xD) ----
#pragma unroll
    for (int nt = 0; nt < 8; nt++) {
#pragma unroll
      for (int kk = 0; kk < 4; kk++) {
        int ka = kk * 4 + 2 * half;
        v2f a, bb;
        a.x  = sS[col][ka];  a.y = sS[col][ka + 1];
        bb.x = Vb[(jt * 16 + ka)     * D + nt * 16 + col];
        bb.y = Vb[(jt * 16 + ka + 1) * D + nt * 16 + col];
        out[nt] = __builtin_amdgcn_wmma_f32_16x16x4_f32(false, a, false, bb, (short)0,
                                                        out[nt], false, false);
      }
    }
  }
  // ---- epilogue: ret[l,d] /= sqrt((l+1)*K) ----
#pragma unroll
  for (int nt = 0; nt < 8; nt++) {
#pragma unroll
    for (int r = 0; r < 8; r++) {
      int row = qt * 16 + r + 8 * half;
      float sc = rsqrtf((float)(row + 1) * (float)Korig);
      rb[row * D + nt * 16 + col] = out[nt][r] * sc;
    }
  }
}

// ---------------- pos block: chunked scan over L -----------------------------
__global__ void k_pos_partial(const float* __restrict__ v, const float* __restrict__ ph,
                              float* __restrict__ csum, int L, int D, int chunk) {
  int b = blockIdx.y, ch = blockIdx.x, d = threadIdx.x;
  float sr = 0.0f, si = 0.0f;
  for (int l = ch * chunk; l < (ch + 1) * chunk; l++) {
    float p = ph[l * D + d], vv = v[(b * L + l) * D + d];
    sr += vv * cosf(p);  si += vv * sinf(p);
  }
  csum[((b * gridDim.x + ch) * 2 + 0) * D + d] = sr;
  csum[((b * gridDim.x + ch) * 2 + 1) * D + d] = si;
}
__global__ void k_pos_final(const float* __restrict__ v, const float* __restrict__ ph,
                            const float* __restrict__ csum, float* __restrict__ ret,
                            int L, int D, int chunk, float invsqrtD) {
  int b = blockIdx.y, ch = blockIdx.x, d = threadIdx.x;
  float mr = 0.0f, mi = 0.0f;
  for (int c = 0; c < ch; c++) {
    mr += csum[((b * gridDim.x + c) * 2 + 0) * D + d];
    mi += csum[((b * gridDim.x + c) * 2 + 1) * D + d];
  }
  for (int l = ch * chunk; l < (ch + 1) * chunk; l++) {
    float p = ph[l * D + d];
    float c_ = cosf(p), s_ = sinf(p);
    mr += v[(b * L + l) * D + d] * c_;
    mi += v[(b * L + l) * D + d] * s_;
    ret[(b * L + l) * D + d] = (mr * c_ + mi * s_) * invsqrtD;
  }
}

// ---------------------------------------------------------------------------
extern "C" void kernel_launch(void* const* d_in, const int* in_sizes, int n_in,
                              void* d_out, int out_size, void* d_ws, size_t ws_size,
                              hipStream_t stream) {
  (void)in_sizes; (void)n_in; (void)out_size; (void)ws_size;
  const int B = 2, L = 1024, D = 128, K = 64, VOC = 32000;
  const int M = B * L;                                   // 2048 rows

  auto F = [&](int i) { return (const float*)d_in[i]; };
  const int* tokens = (const int*)d_in[0];
  const float* emb  = F(1);
  const float* ng[4] = { F(2), F(3), F(4), F(5) };
  const float* nb[4] = { F(6), F(7), F(8), F(9) };
  // pos params at 10 + 7*pi : base_phases, v_W, v_b, o_ln_g, o_ln_b, o_W, o_b
  // content params at 24 + 16*ci : k1_W,k1_b,k2_W,k2_b,q1_W,q1_b,q2_W,q2_b,
  //                                a_W,a_b,v_W,v_b,o_ln_g,o_ln_b,o_W,o_b
  const float* nog = F(56); const float* nob = F(57);
  const float* hW  = F(58); const float* hb  = F(59);

  float* ws = (float*)d_ws;
  const size_t S = (size_t)M * D;                        // 262144 floats
  float* hA  = ws;            float* hB  = ws + S;
  float* xn  = ws + 2 * S;    float* rn  = ws + 3 * S;
  float* t0  = ws + 4 * S;    float* t1  = ws + 5 * S;   // h1 / V
  float* Qc  = ws + 6 * S;    float* Kc  = ws + 7 * S;
  float* ret = ws + 8 * S;
  float* kp  = ws + 9 * S;    float* qp  = kp + S / 2;   // (M*K) each
  float* amp = ws + 10 * S;   float* csum = amp + S / 2; // B*8*2*D

  auto lin = [&](const float* X, const float* W, const float* bias, const float* res,
                 float* Y, int Mm, int N, int Kd, int act) {
    k_linear<<<(Mm / 16) * (N / 16), 32, 0, stream>>>(X, W, bias, res, Y, N, Kd, act);
  };

  k_tdm_nop<<<1, 32, 0, stream>>>();                     // NULL-tensor TDM op
  k_embed<<<M, D, 0, stream>>>(tokens, emb, hA);

  float* cur = hA; float* nxt = hB;
  for (int i = 0; i < 4; i++) {
    k_layernorm<<<M, 128, 0, stream>>>(cur, ng[i], nb[i], xn);
    if ((i & 1) == 0) {                                  // --- pos block ---
      int base = 10 + 7 * (i / 2);
      const float* phs = F(base + 0);
      lin(xn, F(base + 1), F(base + 2), nullptr, t0, M, D, D, 0);        // v
      k_pos_partial<<<dim3(8, B), 128, 0, stream>>>(t0, phs, csum, L, D, 128);
      k_pos_final<<<dim3(8, B), 128, 0, stream>>>(t0, phs, csum, ret, L, D, 128,
                                                  0.08838834764831845f); // 1/sqrt(128)
      k_layernorm<<<M, 128, 0, stream>>>(ret, F(base + 3), F(base + 4), rn);
      lin(rn, F(base + 5), F(base + 6), cur, nxt, M, D, D, 0);           // residual
    } else {                                             // --- content block ---
      int base = 24 + 16 * ((i - 1) / 2);
      lin(xn, F(base + 0), F(base + 1), nullptr, t0, M, D, D, 1);        // gelu
      lin(t0, F(base + 2), F(base + 3), nullptr, kp, M, K, D, 2);        // tanh*pi
      lin(xn, F(base + 4), F(base + 5), nullptr, t0, M, D, D, 1);
      lin(t0, F(base + 6), F(base + 7), nullptr, qp, M, K, D, 2);
      lin(xn, F(base + 8), F(base + 9), nullptr, amp, M, K, D, 3);       // softplus+0.1
      lin(xn, F(base + 10), F(base + 11), nullptr, t1, M, D, D, 0);      // V
      k_phasor<<<(M * K + 255) / 256, 256, 0, stream>>>(kp, qp, amp, Kc, Qc, M * K, K);
      k_attn<<<dim3(L / 16, B), 32, 0, stream>>>(Qc, Kc, t1, ret, L, 2 * K, D, K);
      k_layernorm<<<M, 128, 0, stream>>>(ret, F(base + 12), F(base + 13), rn);
      lin(rn, F(base + 14), F(base + 15), cur, nxt, M, D, D, 0);         // residual
    }
    float* tmp = cur; cur = nxt; nxt = tmp;
  }

  k_layernorm<<<M, 128, 0, stream>>>(cur, nog, nob, xn);
  k_head<<<dim3(M / 64, VOC / 64), 128, 0, stream>>>(xn, hW, hb, (float*)d_out, VOC);
}